// GHMS_Loss_46686294508030
// MI455X (gfx1250) — compile-verified
//
#include <hip/hip_runtime.h>
#include <math.h>

#define BS    1048576
#define CLA   32
#define NBINS 10
#define TOTF  33554432.0f   // BS*CLA
#define NPART 1024          // k3 block count; multiple of 64 for WMMA reducer

typedef __attribute__((ext_vector_type(2))) float v2f;
typedef __attribute__((ext_vector_type(8))) float v8f;

// Workspace layout (bytes):
//   [0,64)                 : unsigned counts[10] (padded)
//   [64,128)               : float wfinal[10]
//   [128,128+4*NPART)      : float partials[NPART]
//   [4352, 4352+4*BS)      : float bce_mean[BS]
//   [4352+4*BS, +BS)       : uchar bin_idx[BS]
#define WS_COUNTS 0
#define WS_WFINAL 64
#define WS_PART   128
#define WS_BCE    4352
#define WS_IDX    (4352 + BS * 4)

__global__ void k0_init(unsigned* __restrict__ counts) {
    if (threadIdx.x < NBINS) counts[threadIdx.x] = 0u;
}

__device__ __forceinline__ void bce_el(float x, float t, float& a, float& b) {
    a = fabsf(x - t);
    // targets are exactly 0.0 or 1.0 -> one transcendental per element
    b = (t > 0.5f) ? (-logf(x)) : (-log1pf(-x));
}

// Pass 1: per-row g and BCE mean in one fully-coalesced sweep (256 MB read once).
__global__ __launch_bounds__(256) void k1_rowstats(
        const float* __restrict__ inp, const float* __restrict__ tgt,
        float* __restrict__ bce_mean, unsigned char* __restrict__ bin_idx,
        unsigned* __restrict__ counts) {
    __shared__ unsigned hist[NBINS];
    if (threadIdx.x < NBINS) hist[threadIdx.x] = 0u;
    __syncthreads();

    const int lane   = threadIdx.x & 31;
    const int wave   = (blockIdx.x * blockDim.x + threadIdx.x) >> 5;
    const int nwaves = (gridDim.x * blockDim.x) >> 5;
    const int ngroups = BS / 4;                 // 4 rows (512 B/operand) per wave-iter
    const float4* __restrict__ in4 = (const float4*)inp;
    const float4* __restrict__ tg4 = (const float4*)tgt;

    for (int grp = wave; grp < ngroups; grp += nwaves) {
        const int base  = grp * 32 + lane;      // float4 index, contiguous per wave
        const int nbase = base + nwaves * 32;   // next tile for this wave
        if (nbase < ngroups * 32) {             // CDNA5 global_prefetch_b8
            __builtin_prefetch(&in4[nbase], 0, 1);
            __builtin_prefetch(&tg4[nbase], 0, 1);
        }
        float4 x = in4[base];
        float4 t = tg4[base];
        float a0, b0, a1, b1, a2, b2, a3, b3;
        bce_el(x.x, t.x, a0, b0);
        bce_el(x.y, t.y, a1, b1);
        bce_el(x.z, t.z, a2, b2);
        bce_el(x.w, t.w, a3, b3);
        float a = (a0 + a1) + (a2 + a3);
        float b = (b0 + b1) + (b2 + b3);
        // segmented butterfly: lanes 8r..8r+7 hold row r -> reduce within groups of 8
        a += __shfl_xor(a, 1, 32);  b += __shfl_xor(b, 1, 32);
        a += __shfl_xor(a, 2, 32);  b += __shfl_xor(b, 2, 32);
        a += __shfl_xor(a, 4, 32);  b += __shfl_xor(b, 4, 32);
        if ((lane & 7) == 0) {
            const int row = grp * 4 + (lane >> 3);
            const float g = a * (1.0f / 32.0f);
            int bi = (int)(g * 10.0f);          // g >= 0 -> trunc == floor
            bi = bi > (NBINS - 1) ? (NBINS - 1) : bi;
            bce_mean[row] = b * (1.0f / 32.0f);
            bin_idx[row]  = (unsigned char)bi;
            atomicAdd(&hist[bi], 1u);
        }
    }
    __syncthreads();
    if (threadIdx.x < NBINS) atomicAdd(&counts[threadIdx.x], hist[threadIdx.x]);
}

// Pass 2 (tiny): per-bin final weights. sum(w) = sum_b counts[b]*pbw[b]^2.
__global__ void k2_weights(const unsigned* __restrict__ counts,
                           float* __restrict__ wfinal) {
    __shared__ float pbw2[NBINS];
    __shared__ float cw[NBINS];
    __shared__ float sumw;
    const int b = threadIdx.x;
    if (b < NBINS) {
        const float s   = (b == 0) ? 1.0f : ((b == 1) ? 0.75f : 0.0f); // smooth[]
        const float cnt = (float)counts[b];
        float pbw = 0.0f;
        if (cnt > 0.0f && s > 0.0f)
            pbw = s * TOTF / (0.25f * cnt);     // acc_sum = (1-momentum)*counts
        pbw2[b] = pbw * pbw;
        cw[b]   = cnt * pbw2[b];
    }
    __syncthreads();
    if (b == 0) {
        float acc = 0.0f;
        for (int i = 0; i < NBINS; ++i) acc += cw[i];
        sumw = acc;
    }
    __syncthreads();
    if (b < NBINS) {
        const float wn = pbw2[b] / sumw;        // matches reference 0-div semantics
        wfinal[b] = (wn < 1e-6f) ? 0.0f : wn;
    }
}

// Pass 3: weighted sum over rows (5 MB read), fixed-order per-block reduction.
__global__ __launch_bounds__(256) void k3_partial(
        const float* __restrict__ bce_mean, const unsigned char* __restrict__ bin_idx,
        const float* __restrict__ wfinal, float* __restrict__ partials) {
    __shared__ float wf[NBINS];
    __shared__ float wsum[8];
    if (threadIdx.x < NBINS) wf[threadIdx.x] = wfinal[threadIdx.x];
    __syncthreads();
    float acc = 0.0f;
    const int tid    = blockIdx.x * blockDim.x + threadIdx.x;
    const int stride = gridDim.x * blockDim.x;
    for (int r = tid; r < BS; r += stride)
        acc += bce_mean[r] * wf[bin_idx[r]];
    acc += __shfl_xor(acc, 1, 32);
    acc += __shfl_xor(acc, 2, 32);
    acc += __shfl_xor(acc, 4, 32);
    acc += __shfl_xor(acc, 8, 32);
    acc += __shfl_xor(acc, 16, 32);
    if ((threadIdx.x & 31) == 0) wsum[threadIdx.x >> 5] = acc;
    __syncthreads();
    if (threadIdx.x == 0) {
        float s = 0.0f;
        for (int w = 0; w < 8; ++w) s += wsum[w];
        partials[blockIdx.x] = s;
    }
}

// Pass 4: one wave; WMMA f32 16x16x4 with ones-B as an accumulating 64-way adder.
// C accumulates across iterations; row-sum m lands in C-vgpr m%8 at lanes 0..15
// (m<8) / 16..31 (m>=8); grand total = sum of 8 C regs at lane0 + lane16.
__global__ void k4_final(const float* __restrict__ partials, float* __restrict__ out) {
    const int lane = threadIdx.x;   // launched with exactly 32 threads, EXEC all-1s
    v2f ones; ones[0] = 1.0f; ones[1] = 1.0f;
    v8f c = {};
    for (int i = 0; i < NPART; i += 64) {
        v2f a;
        a[0] = partials[i + lane];
        a[1] = partials[i + 32 + lane];
        c = __builtin_amdgcn_wmma_f32_16x16x4_f32(
                /*neg_a=*/false, a, /*neg_b=*/false, ones,
                /*c_mod=*/(short)0, c, /*reuse_a=*/false, /*reuse_b=*/false);
    }
    float s = ((c[0] + c[1]) + (c[2] + c[3])) + ((c[4] + c[5]) + (c[6] + c[7]));
    s += __shfl_xor(s, 16, 32);
    if (lane == 0) out[0] = s;
}

extern "C" void kernel_launch(void* const* d_in, const int* in_sizes, int n_in,
                              void* d_out, int out_size, void* d_ws, size_t ws_size,
                              hipStream_t stream) {
    const float* inp = (const float*)d_in[0];
    const float* tgt = (const float*)d_in[1];
    float* out = (float*)d_out;
    char* ws = (char*)d_ws;

    unsigned*      counts   = (unsigned*)(ws + WS_COUNTS);
    float*         wfinal   = (float*)(ws + WS_WFINAL);
    float*         partials = (float*)(ws + WS_PART);
    float*         bce_mean = (float*)(ws + WS_BCE);
    unsigned char* bin_idx  = (unsigned char*)(ws + WS_IDX);

    k0_init<<<1, 32, 0, stream>>>(counts);
    k1_rowstats<<<2048, 256, 0, stream>>>(inp, tgt, bce_mean, bin_idx, counts);
    k2_weights<<<1, 32, 0, stream>>>(counts, wfinal);
    k3_partial<<<NPART, 256, 0, stream>>>(bce_mean, bin_idx, wfinal, partials);
    k4_final<<<1, 32, 0, stream>>>(partials, out);
}